// MyModelONNX_43662637531518
// MI455X (gfx1250) — compile-verified
//
#include <hip/hip_runtime.h>

// Swin shifted-window partition:
//   in : (8, 222, 222, 256) f32
//   out: (8*32*32, 49, 256) f32   (pad 222->224, roll(-3,-3), 7x7 windows)
//
// Pure streaming permute, HBM-bound (~815 MB @ 23.3 TB/s ~ 35us).
// Strategy: one 64-lane group handles one (window, i) pair = 7 output rows
// (7 KB). Index decomposition amortized over 7 rows; inner j-loop is
// add + wrap-select + b128 NT load + b128 NT store per 1 KB row.

typedef float v4f __attribute__((ext_vector_type(4)));

#define WSZ    7
#define SHIFT  3
#define CH     256
#define NB     8
#define HIN    222
#define WIN    222
#define HPAD   224
#define WPAD   224
#define NWH    (HPAD / WSZ)                   // 32
#define NWW    (WPAD / WSZ)                   // 32
#define NWIN   (NB * NWH * NWW)               // 8192 windows
#define NGRP   (NWIN * WSZ)                   // 57344 (window, i) groups

__global__ __launch_bounds__(256) void
swin_window_partition_kernel(const float* __restrict__ q,
                             float* __restrict__ out)
{
    const long long t = (long long)blockIdx.x * blockDim.x + threadIdx.x;
    const int grp  = (int)(t >> 6);    // one (nw, i) pair per 64 lanes
    const int lane = (int)(t & 63);    // float4 index within each 1KB row
    if (grp >= NGRP) return;

    // grp = nw*7 + i ; nw = (b*NWH + wh)*NWW + ww
    const int i  = grp % WSZ;
    const int nw = grp / WSZ;
    const int ww = nw % NWW;
    const int t2 = nw / NWW;
    const int wh = t2 % NWH;
    const int b  = t2 / NWH;

    // source row in original image (undo roll: +SHIFT mod HPAD)
    int hs = wh * WSZ + i + SHIFT; if (hs >= HPAD) hs -= HPAD;
    const bool hok = (hs < HIN);
    const int ws0 = ww * WSZ + SHIFT;

    // base of source image row (units of CH floats), base of dest rows
    const size_t inRowBase = ((size_t)b * HIN + hs) * WIN;
    float* dstBase = out + (((size_t)nw * (WSZ * WSZ)) + (size_t)i * WSZ) * CH
                         + (size_t)lane * 4;

    #pragma unroll
    for (int j = 0; j < WSZ; ++j) {
        int ws = ws0 + j; if (ws >= WPAD) ws -= WPAD;   // undo roll on W
        v4f v = (v4f){0.f, 0.f, 0.f, 0.f};              // pad region -> zeros
        if (hok && ws < WIN) {
            const v4f* src = (const v4f*)(q + (inRowBase + (size_t)ws) * CH) + lane;
            v = __builtin_nontemporal_load(src);        // global_load_b128, TH=NT
        }
        __builtin_nontemporal_store(v, (v4f*)(dstBase + (size_t)j * CH));
    }
}

extern "C" void kernel_launch(void* const* d_in, const int* in_sizes, int n_in,
                              void* d_out, int out_size, void* d_ws, size_t ws_size,
                              hipStream_t stream)
{
    const float* q = (const float*)d_in[0];
    float* out = (float*)d_out;

    // 64 threads per (window, i) group; 256 threads per block -> 4 groups/block
    const long long total_threads = (long long)NGRP * 64;
    const int block = 256;
    const int grid  = (int)((total_threads + block - 1) / block);

    swin_window_partition_kernel<<<grid, block, 0, stream>>>(q, out);
}